// SAGE_60292750902065
// MI455X (gfx1250) — compile-verified
//
#include <hip/hip_runtime.h>
#include <hip/hip_bf16.h>

#define N0v 300000
#define N1v 60000
#define N2v 12000
#define N3v 4096
#define E0v 720000
#define E1v 144000
#define E2v 49152
#define D_INv 300
#define D_HIDv 512
#define D_OUTv 256

typedef __attribute__((ext_vector_type(2))) float v2f;
typedef __attribute__((ext_vector_type(8))) float v8f;

// ---------------- utility kernels ----------------

__global__ void zero_f32(float* __restrict__ p, size_t n) {
    size_t i = (size_t)blockIdx.x * blockDim.x + threadIdx.x;
    size_t stride = (size_t)gridDim.x * blockDim.x;
    for (; i < n; i += stride) p[i] = 0.0f;
}

__global__ void degree_kernel(const int* __restrict__ dst, int E, float* __restrict__ deg) {
    int e = blockIdx.x * blockDim.x + threadIdx.x;
    if (e < E) atomicAdd(&deg[dst[e]], 1.0f);
}

// one thread per (edge, feature): coalesced gather + coalesced atomic scatter
__global__ void scatter_kernel(const float* __restrict__ x,
                               const int* __restrict__ src,
                               const int* __restrict__ dst,
                               long long total, int D,
                               float* __restrict__ agg) {
    long long t = (long long)blockIdx.x * blockDim.x + threadIdx.x;
    if (t >= total) return;
    int e = (int)(t / D);
    int f = (int)(t % D);
    float v = x[(size_t)src[e] * D + f];
    atomicAdd(&agg[(size_t)dst[e] * D + f], v);
}

__global__ void normalize_kernel(float* __restrict__ agg,
                                 const float* __restrict__ deg,
                                 long long total, int D) {
    long long t = (long long)blockIdx.x * blockDim.x + threadIdx.x;
    if (t >= total) return;
    int n = (int)(t / D);
    agg[t] = agg[t] / fmaxf(deg[n], 1.0f);
}

// ---------------- fused dual-GEMM via V_WMMA_F32_16X16X4_F32 ----------------
// out[M,N] = opt_relu( A1[M,K]@W1[K,N] + A2[M,K]@W2[K,N] + bias[N] )
// One wave32 computes one 16x16 tile. K stepped by 4.
// A fragment (16x4 f32, 2 VGPRs): lanes 0-15 row M=lane, K={k,k+1};
//                                 lanes 16-31 row M=lane-16, K={k+2,k+3}.
// B fragment (4x16 f32, 2 VGPRs): lanes 0-15 col N=lane, K rows {k,k+1};
//                                 lanes 16-31 col N=lane-16, K rows {k+2,k+3}.
// C/D (8 VGPRs): vgpr r, lanes 0-15 -> (M=r, N=lane); lanes 16-31 -> (M=r+8, N=lane-16).
__global__ void sage_gemm_wmma(const float* __restrict__ A1, int lda1,
                               const float* __restrict__ A2, int lda2,
                               const float* __restrict__ W1,
                               const float* __restrict__ W2,
                               const float* __restrict__ bias,
                               float* __restrict__ out,
                               float* __restrict__ out_dup,
                               int M, int N, int K, int relu) {
    const int lane = threadIdx.x & 31;
    const int wave = threadIdx.x >> 5;
    const int wavesPerBlock = blockDim.x >> 5;
    const int nTiles = N >> 4;
    const int totalTiles = (M >> 4) * nTiles;
    int tile = blockIdx.x * wavesPerBlock + wave;   // wave-uniform -> EXEC stays full
    if (tile >= totalTiles) return;
    const int mTile = tile / nTiles;
    const int nTile = tile % nTiles;
    const int half = lane >> 4;   // which K-pair this lane carries
    const int l16  = lane & 15;

    v8f c = {};
    for (int pass = 0; pass < 2; ++pass) {
        const float* A = pass ? A2 : A1;
        const int lda  = pass ? lda2 : lda1;
        const float* W = pass ? W2 : W1;
        const float* Arow = A + (size_t)(mTile * 16 + l16) * lda + 2 * half;
        const float* Wcol = W + (nTile * 16 + l16) + (size_t)(2 * half) * N;
        for (int k = 0; k < K; k += 4) {
            v2f a, b;
            a.x = Arow[k];
            a.y = Arow[k + 1];
            b.x = Wcol[(size_t)k * N];
            b.y = Wcol[(size_t)(k + 1) * N];
            c = __builtin_amdgcn_wmma_f32_16x16x4_f32(
                    /*neg_a=*/false, a, /*neg_b=*/false, b,
                    /*c_mod=*/(short)0, c, /*reuse_a=*/false, /*reuse_b=*/false);
        }
    }

    const int outCol = nTile * 16 + l16;
    const float bv = bias[outCol];
#pragma unroll
    for (int r = 0; r < 8; ++r) {
        const int outRow = mTile * 16 + r + 8 * half;
        float v = c[r] + bv;
        if (relu) v = fmaxf(v, 0.0f);
        out[(size_t)outRow * N + outCol] = v;
        if (out_dup) out_dup[(size_t)outRow * N + outCol] = v;
    }
}

// ---------------- host-side orchestration ----------------

static inline unsigned blocksFor(long long total, int bs) {
    return (unsigned)((total + bs - 1) / bs);
}

extern "C" void kernel_launch(void* const* d_in, const int* in_sizes, int n_in,
                              void* d_out, int out_size, void* d_ws, size_t ws_size,
                              hipStream_t stream) {
    const float* x    = (const float*)d_in[0];
    const int* src0   = (const int*)d_in[1];
    const int* dst0   = (const int*)d_in[2];
    const int* src1   = (const int*)d_in[3];
    const int* dst1   = (const int*)d_in[4];
    const int* src2   = (const int*)d_in[5];
    const int* dst2   = (const int*)d_in[6];
    const float* Wl0  = (const float*)d_in[7];
    const float* Wr0  = (const float*)d_in[8];
    const float* b0   = (const float*)d_in[9];
    const float* Wl1  = (const float*)d_in[10];
    const float* Wr1  = (const float*)d_in[11];
    const float* b1   = (const float*)d_in[12];
    const float* Wl2  = (const float*)d_in[13];
    const float* Wr2  = (const float*)d_in[14];
    const float* b2   = (const float*)d_in[15];

    float* ws  = (float*)d_ws;
    float* agg = ws;                                   // up to N1*300 = 18,000,000 floats
    float* deg = ws + (size_t)N1v * D_INv;             // up to N1 floats
    float* h1  = deg + (size_t)N1v;                    // N1*512 = 30,720,000 floats

    float* outp = (float*)d_out;
    float* h2a  = outp;                                // 4096*256
    float* h2b  = outp + (size_t)N3v * D_OUTv;         // 4096*256
    float* pre  = h2b + (size_t)N3v * D_OUTv;          // 12000*512 (relu'd layer-1 out)

    const int BS = 256;

    // ---------- Layer 0: x (N0,300) -> h1 (N1,512), relu ----------
    {
        size_t zn = (size_t)N1v * D_INv + N1v;         // agg + deg contiguous
        zero_f32<<<8192, BS, 0, stream>>>(ws, zn);
        degree_kernel<<<blocksFor(E0v, BS), BS, 0, stream>>>(dst0, E0v, deg);
        long long tot = (long long)E0v * D_INv;
        scatter_kernel<<<blocksFor(tot, BS), BS, 0, stream>>>(x, src0, dst0, tot, D_INv, agg);
        long long ntot = (long long)N1v * D_INv;
        normalize_kernel<<<blocksFor(ntot, BS), BS, 0, stream>>>(agg, deg, ntot, D_INv);
        int tiles = (N1v / 16) * (D_HIDv / 16);
        sage_gemm_wmma<<<blocksFor(tiles, 8), BS, 0, stream>>>(
            agg, D_INv, x, D_INv, Wl0, Wr0, b0, h1, nullptr,
            N1v, D_HIDv, D_INv, /*relu=*/1);
    }

    // ---------- Layer 1: h1 (N1,512) -> pre (N2,512), relu ----------
    {
        zero_f32<<<8192, BS, 0, stream>>>(agg, (size_t)N2v * D_HIDv);
        zero_f32<<<64, BS, 0, stream>>>(deg, (size_t)N2v);
        degree_kernel<<<blocksFor(E1v, BS), BS, 0, stream>>>(dst1, E1v, deg);
        long long tot = (long long)E1v * D_HIDv;
        scatter_kernel<<<blocksFor(tot, BS), BS, 0, stream>>>(h1, src1, dst1, tot, D_HIDv, agg);
        long long ntot = (long long)N2v * D_HIDv;
        normalize_kernel<<<blocksFor(ntot, BS), BS, 0, stream>>>(agg, deg, ntot, D_HIDv);
        int tiles = (N2v / 16) * (D_HIDv / 16);
        sage_gemm_wmma<<<blocksFor(tiles, 8), BS, 0, stream>>>(
            agg, D_HIDv, h1, D_HIDv, Wl1, Wr1, b1, pre, nullptr,
            N2v, D_HIDv, D_HIDv, /*relu=*/1);
    }

    // ---------- Layer 2: pre (N2,512) -> h2 (N3,256), no relu, dual store ----------
    {
        zero_f32<<<8192, BS, 0, stream>>>(agg, (size_t)N3v * D_HIDv);
        zero_f32<<<16, BS, 0, stream>>>(deg, (size_t)N3v);
        degree_kernel<<<blocksFor(E2v, BS), BS, 0, stream>>>(dst2, E2v, deg);
        long long tot = (long long)E2v * D_HIDv;
        scatter_kernel<<<blocksFor(tot, BS), BS, 0, stream>>>(pre, src2, dst2, tot, D_HIDv, agg);
        long long ntot = (long long)N3v * D_HIDv;
        normalize_kernel<<<blocksFor(ntot, BS), BS, 0, stream>>>(agg, deg, ntot, D_HIDv);
        int tiles = (N3v / 16) * (D_OUTv / 16);
        sage_gemm_wmma<<<blocksFor(tiles, 8), BS, 0, stream>>>(
            agg, D_HIDv, pre, D_HIDv, Wl2, Wr2, b2, h2a, h2b,
            N3v, D_OUTv, D_HIDv, /*relu=*/0);
    }
}